// RAN_72567767433723
// MI455X (gfx1250) — compile-verified
//
#include <hip/hip_runtime.h>
#include <cstdint>
#include <cmath>

#define BB 128
#define AA 512
#define NN 256
#define ROWS_PER_BLOCK 32
#define BLOCKS_PER_BATCH (AA / ROWS_PER_BLOCK) /* 16 */
#define TBL 90
#define NEG_005_E (-0.13591409142295226f) /* -0.05 * e */

typedef int   v4i __attribute__((ext_vector_type(4)));
typedef float v4f __attribute__((ext_vector_type(4)));

__global__ __launch_bounds__(256) void RAN_72567767433723_kernel(
    const float* __restrict__ atomic_numbers,  // (B, A)
    const int*   __restrict__ neighbors,       // (B, A, N)
    const float* __restrict__ r_ij,            // (B, A, N)
    float* __restrict__ out)                   // (B, A, N)
{
    __shared__ float an_row[AA];          // staged atomic numbers (2 KB)
    __shared__ int   an_int[AA];          // integer Z per atom (2 KB)
    __shared__ float table[TBL * TBL];    // exp(...) LUT over (zi, zj) (32.4 KB)

    const int tid  = threadIdx.x;
    const int bid  = blockIdx.x;
    const int b    = bid / BLOCKS_PER_BATCH;
    const int row0 = (bid % BLOCKS_PER_BATCH) * ROWS_PER_BLOCK;

    // ---- 1) Async global->LDS DMA of atomic_numbers[b, :] (2 KB).
    // 256 lanes x 4B per instruction; two instructions cover 2048 bytes.
    // ASYNCcnt-tracked; overlaps the transcendental LUT build below.
    {
        unsigned lds_off = (unsigned)(uintptr_t)(&an_row[tid]);  // low 32 bits = LDS byte offset
        unsigned goff    = (unsigned)(tid * 4);                  // byte offset into the row
        const float* src = atomic_numbers + (size_t)b * AA;
        asm volatile(
            "global_load_async_to_lds_b32 %0, %1, %2\n\t"
            "global_load_async_to_lds_b32 %0, %1, %2 offset:1024"
            :
            : "v"(lds_off), "v"(goff), "s"(src)
            : "memory");
    }

    // ---- 2) Build the 90x90 pair LUT while the DMA is in flight.
    // dist_edit depends only on (zi, zj), zi,zj in [1,89] -> 7921 distinct values.
    for (int idx = tid; idx < TBL * TBL; idx += 256) {
        const int zi = idx / TBL;
        const int zj = idx % TBL;
        float v = 0.0f;
        if (zi > 0 && zj > 0) {
            const float p = (float)(zi * zj);
            const float s = (float)(zi + zj);
            v = expf(NEG_005_E * powf(p, 1.01f) / powf(s, 1.1f));
        }
        table[idx] = v;
    }

    // ---- 3) Fence the async copy, make it workgroup-visible.
    asm volatile("s_wait_asynccnt 0" ::: "memory");
    __syncthreads();

    // ---- 3b) One-time float->int conversion so the hot loop has no v_cvt.
    an_int[tid]       = (int)an_row[tid];
    an_int[tid + 256] = (int)an_row[tid + 256];
    __syncthreads();

    // ---- 4) Streaming loop: 8 passes x 4 rows; 64 lanes x 4-wide vectors per row.
    const int lane = tid & 63;   // position within row (x4 elements)
    const int rsub = tid >> 6;   // 0..3: which row of this pass

    for (int p = 0; p < 8; ++p) {
        const int    a    = row0 + p * 4 + rsub;
        const int    zi   = an_int[a];
        const float* tz   = &table[zi * TBL];
        const size_t base = (((size_t)b * AA + a) * NN) + (size_t)(lane * 4);

        if (p < 7) {
            // prefetch next pass's cachelines (global_prefetch_b8)
            const size_t nbase = (((size_t)b * AA + a + 4) * NN) + (size_t)(lane * 4);
            __builtin_prefetch(neighbors + nbase, 0, 0);
            __builtin_prefetch(r_ij + nbase, 0, 0);
        }

        // Non-temporal streaming loads: zero reuse, keep them out of L2 residency.
        const v4i nb = __builtin_nontemporal_load((const v4i*)(neighbors + base));
        const v4f r  = __builtin_nontemporal_load((const v4f*)(r_ij + base));

        v4f o;
        o.x = r.x + tz[an_int[nb.x]];
        o.y = r.y + tz[an_int[nb.y]];
        o.z = r.z + tz[an_int[nb.z]];
        o.w = r.w + tz[an_int[nb.w]];

        __builtin_nontemporal_store(o, (v4f*)(out + base));
    }
}

extern "C" void kernel_launch(void* const* d_in, const int* in_sizes, int n_in,
                              void* d_out, int out_size, void* d_ws, size_t ws_size,
                              hipStream_t stream) {
    (void)in_sizes; (void)n_in; (void)out_size; (void)d_ws; (void)ws_size;
    const float* atomic_numbers = (const float*)d_in[0];
    const int*   neighbors      = (const int*)d_in[1];
    const float* r_ij           = (const float*)d_in[2];
    float*       out            = (float*)d_out;

    dim3 grid(BB * BLOCKS_PER_BATCH);  // 2048 blocks, each fully within one batch
    dim3 block(256);                   // 8 wave32 waves
    RAN_72567767433723_kernel<<<grid, block, 0, stream>>>(atomic_numbers, neighbors, r_ij, out);
}